// get_enhanced_source_68186900791975
// MI455X (gfx1250) — compile-verified
//
#include <hip/hip_runtime.h>
#include <hip/hip_bf16.h>
#include <math.h>

// Problem constants (from reference): B=16, H=W=38 -> L=1444, A=4, NC=21, C=64
#define B_   16
#define L_   1444          // 38*38
#define AN_  4
#define NC_  21
#define C_   64
#define KC_  76            // K-chunk: 1444 = 19 * 76, divisible by WMMA K=4
#define LDP_ 65            // padded LDS row stride for the src stage

typedef __attribute__((ext_vector_type(2))) float v2f;
typedef __attribute__((ext_vector_type(8))) float v8f;

// ---------------------------------------------------------------------------
// Kernel 1: per-(b,l) grouped softmax + argmax%NC  -> cls_idx  (int, in ws)
// ---------------------------------------------------------------------------
__global__ void classify_kernel(const float* __restrict__ cls_pred,
                                int* __restrict__ cls_idx) {
    int idx = blockIdx.x * blockDim.x + threadIdx.x;
    if (idx >= B_ * L_) return;
    const float* p = cls_pred + (size_t)idx * (AN_ * NC_);
    float best = -3.4e38f;
    int bestg = 0;
    for (int a = 0; a < AN_; ++a) {
        const float* q = p + a * NC_;
        float m = q[0];
        for (int j = 1; j < NC_; ++j) m = fmaxf(m, q[j]);
        float s = 0.0f;
        for (int j = 0; j < NC_; ++j) s += expf(q[j] - m);
        float inv = 1.0f / s;
        for (int j = 0; j < NC_; ++j) {
            float prob = expf(q[j] - m) * inv;   // matches jax softmax values
            if (prob > best) { best = prob; bestg = a * NC_ + j; }  // first-max wins
        }
    }
    cls_idx[idx] = bestg % NC_;
}

// ---------------------------------------------------------------------------
// Kernel 2: NCHW -> (B, L, C) transpose via LDS tile.
// Writes directly into the second output region (the `src` return value),
// which is also the GEMM B-matrix in row-major (j, c) layout.
// ---------------------------------------------------------------------------
__global__ void transpose_kernel(const float* __restrict__ source,
                                 float* __restrict__ src) {
    __shared__ float tile[64][C_ + 1];
    const int b  = blockIdx.y;
    const int l0 = blockIdx.x * 64;
    const int tx = threadIdx.x & 63;   // 0..63
    const int ty = threadIdx.x >> 6;   // 0..3
    // coalesced read along l from source[b][c][l0+tx]
    for (int c = ty; c < C_; c += 4) {
        int l = l0 + tx;
        float v = 0.0f;
        if (l < L_) v = source[((size_t)b * C_ + c) * L_ + l];
        tile[tx][c] = v;
    }
    __syncthreads();
    // coalesced write along c into src[b][l][c]
    for (int i = ty; i < 64; i += 4) {
        int l = l0 + i;
        if (l < L_) src[((size_t)b * L_ + l) * C_ + tx] = tile[i][tx];
    }
}

// ---------------------------------------------------------------------------
// Kernel 3: fused  G(cls_idx, R) @ src  using V_WMMA_F32_16X16X4_F32.
// G is never materialized: each wave generates its 16x4 A-fragment from
// cls_idx (LDS) + the 21x21 table (LDS), fully branchlessly (address select
// via v_cndmask, single ds_load per element).  4 waves/block share a 76x64
// LDS stage of src.  One wave computes one 16-row x 64-col tile of `fused`.
//
// Fragment layouts (CDNA5 ISA):
//   A 16x4 f32 : a[v]   = A[lane%16][(lane>>4)*2 + v]
//   C/D 16x16  : acc[r] = D[r + (lane>>4)*8][lane%16]
//   B 4x16 f32 : assumed to mirror A's K slotting:
//                b[v]   = Bmat[(lane>>4)*2 + v][lane%16]
// ---------------------------------------------------------------------------
__global__ void __launch_bounds__(128)
fuse_gemm_kernel(const float* __restrict__ src,      // (B, L, C)
                 const int*   __restrict__ cls_idx,  // (B, L)
                 const float* __restrict__ R,        // (NC, NC)
                 float* __restrict__ fused) {        // (B, L, C)
    __shared__ int   sIdx[L_];            // 5776 B
    __shared__ float sR[NC_ * NC_];       // 1764 B
    __shared__ float sB[KC_ * LDP_];      // 19760 B

    const int b    = blockIdx.y;
    const int tid  = threadIdx.x;
    const int wave = tid >> 5;            // 0..3
    const int lane = tid & 31;

    const float* srcB = src + (size_t)b * L_ * C_;

    for (int t = tid; t < L_; t += 128)        sIdx[t] = cls_idx[b * L_ + t];
    for (int t = tid; t < NC_ * NC_; t += 128) sR[t]   = R[t];
    __syncthreads();

    const int  mt        = blockIdx.x * 4 + wave;   // M-tile id, 0..91
    const int  i0        = mt * 16;
    const bool tileValid = (i0 < L_);

    const int m     = lane & 15;
    const int khalf = lane >> 4;                    // 0 or 1
    const int i     = i0 + m;                       // global output row
    int ci = 0;
    if (tileValid) ci = sIdx[(i < L_) ? i : (L_ - 1)];
    const int   ciNC = ci * NC_;
    const float dg   = (ci != 0) ? 1.0f : 0.0f;     // diagonal value of G

    v8f acc[4];
    for (int s = 0; s < 4; ++s)
        for (int e = 0; e < 8; ++e) acc[s][e] = 0.0f;

    for (int jc = 0; jc < L_; jc += KC_) {
        // stage src rows [jc, jc+KC) into LDS (float4 coalesced)
        for (int t = tid; t < KC_ * C_ / 4; t += 128) {
            int row = (t * 4) / C_;
            int col = (t * 4) % C_;
            const float4 v = *(const float4*)&srcB[(size_t)(jc + row) * C_ + col];
            float* d = &sB[row * LDP_ + col];
            d[0] = v.x; d[1] = v.y; d[2] = v.z; d[3] = v.w;
        }
        // prefetch next K-chunk (global_prefetch_b8 on gfx1250)
        if (jc + KC_ < L_)
            __builtin_prefetch(srcB + (size_t)(jc + KC_) * C_ + tid * 38, 0, 0);
        __syncthreads();

        if (tileValid) {
#pragma unroll 2
            for (int kk = 0; kk < KC_; kk += 4) {
                // ---- A fragment: G[i, j0g .. j0g+1], branchless ----
                const int j0g = jc + kk + khalf * 2;       // even -> 8B aligned
                const int2 cjp = *(const int2*)&sIdx[j0g]; // one ds_load_b64
                v2f a;
#pragma unroll
                for (int v = 0; v < 2; ++v) {
                    const int j  = j0g + v;
                    const int cj = v ? cjp.y : cjp.x;
                    // both addresses always valid -> pure v_cndmask select
                    const int addr = (i > j) ? (ciNC + cj) : (cj * NC_ + ci);
                    float g = sR[addr];
                    a[v] = (i == j) ? dg : g;
                }
                const int brow = kk + khalf * 2;
#pragma unroll
                for (int s = 0; s < 4; ++s) {
                    v2f bb;
                    bb[0] = sB[(brow    ) * LDP_ + s * 16 + m];
                    bb[1] = sB[(brow + 1) * LDP_ + s * 16 + m];
                    acc[s] = __builtin_amdgcn_wmma_f32_16x16x4_f32(
                        false, a, false, bb, (short)0, acc[s], false, false);
                }
            }
        }
        __syncthreads();
    }

    if (tileValid) {
        float* outT = fused + (size_t)b * L_ * C_;
#pragma unroll
        for (int s = 0; s < 4; ++s)
            for (int r = 0; r < 8; ++r) {
                int row = i0 + r + khalf * 8;
                if (row < L_)
                    outT[(size_t)row * C_ + s * 16 + m] = acc[s][r];
            }
    }
}

// ---------------------------------------------------------------------------
extern "C" void kernel_launch(void* const* d_in, const int* in_sizes, int n_in,
                              void* d_out, int out_size, void* d_ws, size_t ws_size,
                              hipStream_t stream) {
    const float* cls_pred   = (const float*)d_in[0];  // (B, 38, 38, 84)
    const float* source     = (const float*)d_in[1];  // (B, 64, 38, 38)
    const float* cls_r_prob = (const float*)d_in[2];  // (21, 21)
    // d_in[3] = num_classes (scalar, hardcoded as NC_=21)

    float* fused = (float*)d_out;                       // (B, L, 64)
    float* src   = (float*)d_out + (size_t)B_ * L_ * C_;// (B, L, 64) second output
    int* cls_idx = (int*)d_ws;                          // B*L ints of scratch

    classify_kernel<<<(B_ * L_ + 255) / 256, 256, 0, stream>>>(cls_pred, cls_idx);
    transpose_kernel<<<dim3((L_ + 63) / 64, B_), 256, 0, stream>>>(source, src);
    fuse_gemm_kernel<<<dim3(23, B_), 128, 0, stream>>>(src, cls_idx, cls_r_prob, fused);
}